// L1loss_19859928776791
// MI455X (gfx1250) — compile-verified
//
#include <hip/hip_runtime.h>

typedef __attribute__((ext_vector_type(4))) float f32x4;
typedef __attribute__((ext_vector_type(2))) float v2f;
typedef __attribute__((ext_vector_type(8))) float v8f;

#define NBLOCKS 2048
#define NTHREADS 256

// Reduce 32 lane-partials to the wave total (replicated in all lanes) using
// V_WMMA_F32_16X16X4_F32:  D = A x B, B = ones  =>  D[m,n] = rowsum_m(A).
// A 16x4 layout (ISA 7.12.2): lane m (0-15) holds A[m,{0,1}], lane m+16 holds
// A[m,{2,3}]. We put (x, 0) per lane, so rowsum_m = x[m] + x[m+16].
// C/D layout: lane L<16 VGPR r = D[r,L]; lane L>=16 VGPR r = D[r+8,L-16].
// Summing a lane's 8 VGPRs gives sum of rowsums 0..7 (low half) or 8..15
// (high half); one shfl_xor(16) add completes the total.
__device__ __forceinline__ float wave_reduce_wmma(float x) {
    v2f a; a.x = x;    a.y = 0.0f;
    v2f b; b.x = 1.0f; b.y = 1.0f;
    v8f c = {};
    c = __builtin_amdgcn_wmma_f32_16x16x4_f32(
        /*neg_a=*/false, a, /*neg_b=*/false, b,
        /*c_mod=*/(short)0, c, /*reuse_a=*/false, /*reuse_b=*/false);
    float s = c[0] + c[1] + c[2] + c[3] + c[4] + c[5] + c[6] + c[7];
    s += __shfl_xor(s, 16, 32);
    return s;
}

// Phase 1: streaming weighted |a-b| reduction, one partial per block.
__global__ __launch_bounds__(NTHREADS) void l1w_partial_kernel(
    const float* __restrict__ inp, const float* __restrict__ tgt,
    const float* __restrict__ w, float* __restrict__ partial,
    int totalV, int d4mask) {
    const f32x4* in4 = (const f32x4*)inp;
    const f32x4* tg4 = (const f32x4*)tgt;
    const f32x4* w4  = (const f32x4*)w;

    int tid    = blockIdx.x * NTHREADS + threadIdx.x;
    int stride = gridDim.x * NTHREADS;

    f32x4 acc = {0.0f, 0.0f, 0.0f, 0.0f};
    for (int v = tid; v < totalV; v += stride) {
        // columns never cross a row boundary: D=4096 is a multiple of 4
        int c4 = v & d4mask;
        f32x4 a  = __builtin_nontemporal_load(in4 + v);  // th:NT, stream once
        f32x4 b  = __builtin_nontemporal_load(tg4 + v);  // th:NT
        f32x4 wv = w4[c4];                               // RT: 16KB, cache-hot
        acc.x = fmaf(wv.x, fabsf(a.x - b.x), acc.x);
        acc.y = fmaf(wv.y, fabsf(a.y - b.y), acc.y);
        acc.z = fmaf(wv.z, fabsf(a.z - b.z), acc.z);
        acc.w = fmaf(wv.w, fabsf(a.w - b.w), acc.w);
    }
    float s = (acc.x + acc.y) + (acc.z + acc.w);

    // wave32 reduction via WMMA, then combine the block's 8 waves in LDS
    float t = wave_reduce_wmma(s);
    __shared__ float wsum[NTHREADS / 32];
    int lane = threadIdx.x & 31;
    int wid  = threadIdx.x >> 5;
    if (lane == 0) wsum[wid] = t;
    __syncthreads();
    if (threadIdx.x == 0) {
        float r = 0.0f;
        #pragma unroll
        for (int i = 0; i < NTHREADS / 32; ++i) r += wsum[i];
        partial[blockIdx.x] = r;
    }
}

// Phase 2: single block folds the NBLOCKS partials and applies 1/B.
__global__ __launch_bounds__(NTHREADS) void l1w_final_kernel(
    const float* __restrict__ partial, float* __restrict__ out,
    int n, float inv_rows) {
    float s = 0.0f;
    for (int i = threadIdx.x; i < n; i += NTHREADS) s += partial[i];

    float t = wave_reduce_wmma(s);
    __shared__ float wsum[NTHREADS / 32];
    int lane = threadIdx.x & 31;
    int wid  = threadIdx.x >> 5;
    if (lane == 0) wsum[wid] = t;
    __syncthreads();
    if (threadIdx.x == 0) {
        float r = 0.0f;
        #pragma unroll
        for (int i = 0; i < NTHREADS / 32; ++i) r += wsum[i];
        out[0] = r * inv_rows;
    }
}

extern "C" void kernel_launch(void* const* d_in, const int* in_sizes, int n_in,
                              void* d_out, int out_size, void* d_ws, size_t ws_size,
                              hipStream_t stream) {
    const float* inp = (const float*)d_in[0];   // [B, D] f32
    const float* tgt = (const float*)d_in[1];   // [B, D] f32
    const float* w   = (const float*)d_in[2];   // [D]    f32
    float* out       = (float*)d_out;           // scalar f32
    float* partial   = (float*)d_ws;            // NBLOCKS floats (8 KB)

    const int D      = in_sizes[2];             // 4096
    const int totalE = in_sizes[0];             // B*D = 2^26
    const int totalV = totalE >> 2;             // float4 count = 2^24
    const int rows   = totalE / D;              // B = 16384
    const int d4mask = (D >> 2) - 1;            // D/4 is a power of two

    l1w_partial_kernel<<<NBLOCKS, NTHREADS, 0, stream>>>(
        inp, tgt, w, partial, totalV, d4mask);
    l1w_final_kernel<<<1, NTHREADS, 0, stream>>>(
        partial, out, NBLOCKS, 1.0f / (float)rows);
}